// SuccessiveOverRelaxation_opt_63385127354926
// MI455X (gfx1250) — compile-verified
//
#include <hip/hip_runtime.h>
#include <stdint.h>

// Grid: 160^3 fp32. Tile: 32(x) x 10(y) x 10(z) interior, halo-2 staged to LDS
// via the Tensor Data Mover, fused red-black SOR sweep per kernel launch.
// Output tile is written back with tensor_store_from_lds (TDM both directions).
#define NXg 160
#define TX 32
#define TY 10
#define TZ 10
#define DXg (TX + 4)   // 36  src tile x (halo 2)
#define DYg (TY + 4)   // 14
#define DZg (TZ + 4)   // 14
#define HXg (TX + 2)   // 34  half tile x (halo 1)
#define HYg (TY + 2)   // 12
#define HZg (TZ + 2)   // 12
#define N_ITERS 120

typedef unsigned int v4u_ __attribute__((ext_vector_type(4)));
typedef int          v8i_ __attribute__((ext_vector_type(8)));
typedef int          v4i_ __attribute__((ext_vector_type(4)));

// Build the 5 descriptor groups for a 3D tile <-> LDS transfer.
// gaddr: global byte address of tile start; ldsoff: LDS byte offset;
// td*: tensor extents from tile start (elements); tile dims fixed per direction.
__device__ __forceinline__ void tdm_desc(unsigned long long gaddr, unsigned ldsoff,
                                         unsigned td0, unsigned td1, unsigned td2,
                                         unsigned tile0, unsigned tile1, unsigned tile2,
                                         v4u_& g0, v8i_& g1, v4i_& g2, v4i_& g3, v8i_& g4) {
  const unsigned s0 = NXg;          // dim0 stride (elements)
  const unsigned s1 = NXg * NXg;    // dim1 stride (elements)
  g0[0] = 1u;                                                  // count=1, user mode
  g0[1] = ldsoff;                                              // lds_addr [63:32]
  g0[2] = (unsigned)gaddr;                                     // global_addr [95:64]
  g0[3] = (unsigned)((gaddr >> 32) & 0x01FFFFFFull)            // global_addr [120:96]
        | (2u << 30);                                          // type=2 ("image") [127:126]
  g1[0] = (int)(2u << 16);                                     // data_size=2 (4B), no multicast
  g1[1] = (int)((td0 & 0xFFFFu) << 16);                        // tensor_dim0[15:0] @bit48
  g1[2] = (int)((td0 >> 16) | ((td1 & 0xFFFFu) << 16));        // dim0 hi | dim1 lo
  g1[3] = (int)((td1 >> 16) | (tile0 << 16));                  // dim1 hi | tile_dim0
  g1[4] = (int)(tile1 | (tile2 << 16));                        // tile_dim1 | tile_dim2
  g1[5] = (int)s0;                                             // dim0_stride[31:0]
  g1[6] = (int)((s1 & 0xFFFFu) << 16);                         // dim0_stride[47:32]=0 | dim1_stride[15:0]
  g1[7] = (int)(s1 >> 16);                                     // dim1_stride[47:16]
  g2[0] = (int)td2;                                            // tensor_dim2
  g2[1] = 0; g2[2] = 0; g2[3] = 0;                             // no dim3 / iterate
  g3[0] = 0; g3[1] = 0; g3[2] = 0; g3[3] = 0;                  // no dim3 stride / dim4
  g4[0] = 0; g4[1] = 0; g4[2] = 0; g4[3] = 0;
  g4[4] = 0; g4[5] = 0; g4[6] = 0; g4[7] = 0;
}

__global__ __launch_bounds__(256)
void sor_init_kernel(const float* __restrict__ img, float* __restrict__ out, int n) {
  int i = blockIdx.x * blockDim.x + threadIdx.x;
  if (i < n) out[i] = 0.5f * img[i] + img[n + i];  // 0.5*ch0 + ch1
}

__global__ __launch_bounds__(256)
void sor_fused_kernel(const float* __restrict__ src, const float* __restrict__ mask,
                      float* __restrict__ dst) {
  __shared__ float s_src[DZg * DYg * DXg];   // 28,224 B
  __shared__ float s_half[HZg * HYg * HXg];  // 19,584 B
  __shared__ float s_out[TZ * TY * TX];      // 12,800 B  (total 60,608 B)

  const int x0 = blockIdx.x * TX;
  const int y0 = blockIdx.y * TY;
  const int z0 = blockIdx.z * TZ;
  const int ox = x0 - 2, oy = y0 - 2, oz = z0 - 2;
  const int lx = ox < 0 ? 0 : ox;   // clamped tile-load origin
  const int ly = oy < 0 ? 0 : oy;
  const int lz = oz < 0 ? 0 : oz;

  // ---- TDM: wave 0 issues a 3D tensor tile load (global -> LDS) ------------
  if (threadIdx.x < 32u) {
    const unsigned long long gaddr =
        (unsigned long long)(uintptr_t)(src + ((size_t)lz * NXg + (size_t)ly) * NXg + (size_t)lx);
    v4u_ g0; v8i_ g1; v4i_ g2; v4i_ g3; v8i_ g4;
    tdm_desc(gaddr, (unsigned)(uintptr_t)(&s_src[0]),
             (unsigned)(NXg - lx), (unsigned)(NXg - ly), (unsigned)(NXg - lz),
             DXg, DYg, DZg, g0, g1, g2, g3, g4);
    __builtin_amdgcn_tensor_load_to_lds(g0, g1, g2, g3, g4, 0);
    __builtin_amdgcn_s_wait_tensorcnt(0);
  }
  __syncthreads();

  const float w  = 2.0f / (1.0f + 3.14159265358979323846f);  // wopt (min_dim == 1)
  const float c0 = 1.0f - w;
  const float cn = w / 6.0f;

#define SRC_AT(gz_, gy_, gx_) \
  s_src[(((gz_) - lz) * DYg + ((gy_) - ly)) * DXg + ((gx_) - lx)]
#define SRC_OR0(gz_, gy_, gx_)                                                     \
  (((unsigned)(gx_) < NXg && (unsigned)(gy_) < NXg && (unsigned)(gz_) < NXg)       \
       ? SRC_AT(gz_, gy_, gx_) : 0.0f)

  // ---- half sweep: update black cells (parity 1) over halo-1 region --------
  for (int t = (int)threadIdx.x; t < HZg * HYg * HXg; t += (int)blockDim.x) {
    const int hx = t % HXg;
    const int r0 = t / HXg;
    const int hy = r0 % HYg;
    const int hz = r0 / HYg;
    const int gx = x0 - 1 + hx;
    const int gy = y0 - 1 + hy;
    const int gz = z0 - 1 + hz;
    float v = 0.0f;
    if ((unsigned)gx < NXg && (unsigned)gy < NXg && (unsigned)gz < NXg) {
      const float c = SRC_AT(gz, gy, gx);
      if (((gx + gy + gz) & 1) == 1) {
        const float nb = SRC_OR0(gz, gy, gx - 1) + SRC_OR0(gz, gy, gx + 1)
                       + SRC_OR0(gz, gy - 1, gx) + SRC_OR0(gz, gy + 1, gx)
                       + SRC_OR0(gz - 1, gy, gx) + SRC_OR0(gz + 1, gy, gx);
        const float conv = fmaf(cn, nb, c0 * c);
        const float m = mask[((size_t)gz * NXg + gy) * NXg + gx];
        v = fmaf(m, conv - c, c);           // m*conv + (1-m)*c
      } else {
        v = c;                              // red cells unchanged at half step
      }
    }
    s_half[t] = v;                          // OOB coords hold 0 (== zero padding)
  }
  __syncthreads();

  // ---- full sweep: update red cells (parity 0), mask-blend vs cur ----------
  for (int t = (int)threadIdx.x; t < TX * TY * TZ; t += (int)blockDim.x) {
    const int ix = t % TX;
    const int r0 = t / TX;
    const int iy = r0 % TY;
    const int iz = r0 / TY;
    const int gx = x0 + ix, gy = y0 + iy, gz = z0 + iz;
    const int hx = ix + 1, hy = iy + 1, hz = iz + 1;
    const float hc = s_half[(hz * HYg + hy) * HXg + hx];
    const float cc = SRC_AT(gz, gy, gx);    // iteration input (aux for blending)
    const float m  = mask[((size_t)gz * NXg + gy) * NXg + gx];
    float out;
    if (((gx + gy + gz) & 1) == 0) {
      const float nb = s_half[(hz * HYg + hy) * HXg + hx - 1]
                     + s_half[(hz * HYg + hy) * HXg + hx + 1]
                     + s_half[(hz * HYg + hy - 1) * HXg + hx]
                     + s_half[(hz * HYg + hy + 1) * HXg + hx]
                     + s_half[((hz - 1) * HYg + hy) * HXg + hx]
                     + s_half[((hz + 1) * HYg + hy) * HXg + hx];
      const float conv = fmaf(cn, nb, c0 * hc);
      out = fmaf(m, conv - cc, cc);         // m*conv(half) + (1-m)*cur
    } else {
      out = fmaf(m, hc - cc, cc);           // m*half + (1-m)*cur
    }
    s_out[t] = out;                         // stage tile; TDM stores it below
  }
  __syncthreads();

  // ---- TDM: wave 0 stores the interior tile (LDS -> global) ----------------
  // S_ENDPGM performs an implicit wait-for-idle, covering TENSORcnt, so the
  // LDS stays live until the DMA completes; no explicit wait needed.
  if (threadIdx.x < 32u) {
    const unsigned long long gaddr =
        (unsigned long long)(uintptr_t)(dst + ((size_t)z0 * NXg + (size_t)y0) * NXg + (size_t)x0);
    v4u_ g0; v8i_ g1; v4i_ g2; v4i_ g3; v8i_ g4;
    tdm_desc(gaddr, (unsigned)(uintptr_t)(&s_out[0]),
             (unsigned)(NXg - x0), (unsigned)(NXg - y0), (unsigned)(NXg - z0),
             TX, TY, TZ, g0, g1, g2, g3, g4);
    __builtin_amdgcn_tensor_store_from_lds(g0, g1, g2, g3, g4, 0);
  }
#undef SRC_AT
#undef SRC_OR0
}

extern "C" void kernel_launch(void* const* d_in, const int* in_sizes, int n_in,
                              void* d_out, int out_size, void* d_ws, size_t ws_size,
                              hipStream_t stream) {
  (void)in_sizes; (void)n_in; (void)out_size; (void)ws_size;
  const float* img  = (const float*)d_in[0];   // (1,3,160,160,160)
  const float* mask = img;                     // channel 0
  const int n = NXg * NXg * NXg;               // 4,096,000 elements per field

  float* W0  = (float*)d_ws;                   // needs 2*n*4 = 32.8 MB workspace
  float* W1  = W0 + n;
  float* out = (float*)d_out;

  sor_init_kernel<<<dim3((n + 255) / 256), dim3(256), 0, stream>>>(img, W0, n);

  dim3 grid(NXg / TX, NXg / TY, NXg / TZ);     // 5 x 16 x 16 = 1280 blocks
  float* cur = W0;
  float* nxt = W1;
  for (int i = 0; i < N_ITERS; ++i) {
    float* dst = (i == N_ITERS - 1) ? out : nxt;
    sor_fused_kernel<<<grid, dim3(256), 0, stream>>>(cur, mask, dst);
    nxt = cur;
    cur = dst;
  }
}